// QuanvLayer1D_27719718928910
// MI455X (gfx1250) — compile-verified
//
#include <hip/hip_runtime.h>

typedef float v2f __attribute__((ext_vector_type(2)));
typedef float v8f __attribute__((ext_vector_type(8)));

#define NQ   8
#define DIM  256            // 2^NQ statevector size
#define KW   4              // conv kernel width
#define PADW 1
#define TWO_PI 6.28318530717958647692f

// ============================================================================
// Precompute: psi0 = U(w0)|0>, and A = U(w2) U(w1) as dense 256x256 complex.
// One block per column (block 256 handles psi0). Statevector lives in LDS,
// thread t owns amplitude t; gates applied with read-barrier-write-barrier.
// Qubit q maps to bit (7-q) of the flat index (row-major (2,)*8 tensor).
// ============================================================================
__device__ __forceinline__ void apply1(float* sr, float* si, int tid, int mask,
    float u00r, float u00i, float u01r, float u01i,
    float u10r, float u10i, float u11r, float u11i) {
  int i0 = tid & ~mask, i1 = tid | mask;
  float v0r = sr[i0], v0i = si[i0], v1r = sr[i1], v1i = si[i1];
  __syncthreads();
  if (tid & mask) {
    sr[tid] = u10r*v0r - u10i*v0i + u11r*v1r - u11i*v1i;
    si[tid] = u10r*v0i + u10i*v0r + u11r*v1i + u11i*v1r;
  } else {
    sr[tid] = u00r*v0r - u00i*v0i + u01r*v1r - u01i*v1i;
    si[tid] = u00r*v0i + u00i*v0r + u01r*v1i + u01i*v1r;
  }
  __syncthreads();
}

__device__ __forceinline__ void cnot_g(float* sr, float* si, int tid, int maskc, int maskt) {
  int src = (tid & maskc) ? (tid ^ maskt) : tid;
  float r = sr[src], im = si[src];
  __syncthreads();
  sr[tid] = r; si[tid] = im;
  __syncthreads();
}

__device__ void apply_ansatz(float* sr, float* si, int tid, const float* w8) {
  for (int j = 0; j < NQ; ++j) {
    int cq = (j < NQ-1) ? j : 0;
    int tq = (j < NQ-1) ? j+1 : NQ-1;
    int mj = 1 << (7 - j);
    int mc = 1 << (7 - cq), mt = 1 << (7 - tq);
    float h = 0.5f * w8[j];
    float c = cosf(h), s = sinf(h);
    apply1(sr, si, tid, mj,  c,0.f,   0.f,-s,   0.f,-s,   c,0.f);   // RX
    cnot_g(sr, si, tid, mc, mt);
    apply1(sr, si, tid, mj,  c,0.f,   -s,0.f,   s,0.f,    c,0.f);   // RY
    cnot_g(sr, si, tid, mc, mt);
    apply1(sr, si, tid, mj,  c,-s,    0.f,0.f,  0.f,0.f,  c,s);     // RZ
  }
}

__global__ void precompute_kernel(const float* __restrict__ weights,
                                  float* __restrict__ Ar, float* __restrict__ Ai,
                                  float* __restrict__ p0r, float* __restrict__ p0i) {
  __shared__ float sr[DIM], si[DIM];
  int tid = threadIdx.x;
  int blk = blockIdx.x;
  if (blk == DIM) {                         // psi0 = U(w0)|0>
    float w0[8];
    for (int j = 0; j < 8; ++j) { float t = fmodf(weights[j], TWO_PI); if (t < 0.f) t += TWO_PI; w0[j] = t; }
    sr[tid] = (tid == 0) ? 1.f : 0.f; si[tid] = 0.f;
    __syncthreads();
    apply_ansatz(sr, si, tid, w0);
    p0r[tid] = sr[tid]; p0i[tid] = si[tid];
  } else {                                  // column blk of A = U(w2)U(w1)
    float w1[8], w2[8];
    for (int j = 0; j < 8; ++j) {
      float t1 = fmodf(weights[8 + j],  TWO_PI); if (t1 < 0.f) t1 += TWO_PI; w1[j] = t1;
      float t2 = fmodf(weights[16 + j], TWO_PI); if (t2 < 0.f) t2 += TWO_PI; w2[j] = t2;
    }
    sr[tid] = (tid == blk) ? 1.f : 0.f; si[tid] = 0.f;
    __syncthreads();
    apply_ansatz(sr, si, tid, w1);
    apply_ansatz(sr, si, tid, w2);
    Ar[tid*DIM + blk] = sr[tid];            // row-major [m][k]
    Ai[tid*DIM + blk] = si[tid];
  }
}

// ============================================================================
// Quanv layer: 16 circuits per workgroup (WMMA N dim), 256 threads = 8 waves.
// psi_enc built per-thread -> LDS; psi_final = A*psi_enc via FP32 WMMA
// (complex as 4 real accumulations); probs -> LDS; signed reductions -> <Z_j>.
// ============================================================================
template<int LAYER>
__global__ void quanv_kernel(const float* __restrict__ src, int Lsrc,
                             const float* __restrict__ Ar, const float* __restrict__ Ai,
                             const float* __restrict__ p0r, const float* __restrict__ p0i,
                             float* __restrict__ out, int Bn) {
  const int PITCH = 17;                     // bank-conflict padding
  __shared__ float XrL[DIM * PITCH];
  __shared__ float XiL[DIM * PITCH];
  __shared__ float PL [DIM * PITCH];
  __shared__ float angs[16 * 4];
  __shared__ float ezs [16 * 8];

  int tid  = threadIdx.x;
  int nseg = Lsrc + 2*PADW - KW + 1;        // Lsrc - 1
  int total = Bn * nseg;
  int base = blockIdx.x * 16;

  // ---- angles for 16 circuits (zero-padded windows) ----
  if (tid < 64) {
    int c = tid >> 2, i = tid & 3;
    int g = base + c;
    float a = 0.f;
    if (g < total) {
      int b = g / nseg, s = g % nseg;
      int pos = s - PADW + i;
      if (pos >= 0 && pos < Lsrc) a = src[b*Lsrc + pos];
    }
    angs[c*4 + i] = a;
  }
  __syncthreads();

  // ---- psi_enc[t] = sum_{m'} RY4[m][m'] * psi0[(m'<<4)|low], m=t>>4 ----
  {
    int m = tid >> 4, low = tid & 15;
    float pr[16], pim[16];
    #pragma unroll
    for (int mp = 0; mp < 16; ++mp) { pr[mp] = p0r[(mp<<4)|low]; pim[mp] = p0i[(mp<<4)|low]; }
    for (int c = 0; c < 16; ++c) {
      float cc[4], ss[4];
      #pragma unroll
      for (int i = 0; i < 4; ++i) { float h = 0.5f*angs[c*4+i]; cc[i] = cosf(h); ss[i] = sinf(h); }
      float aR = 0.f, aI = 0.f;
      for (int mp = 0; mp < 16; ++mp) {
        float coef = 1.f;
        #pragma unroll
        for (int i = 0; i < 4; ++i) {          // RY: [c -s; s c], qubit i = bit (3-i)
          int rb = (m >> (3-i)) & 1, cb = (mp >> (3-i)) & 1;
          float e = (rb == cb) ? cc[i] : (rb ? ss[i] : -ss[i]);
          coef *= e;
        }
        aR += coef * pr[mp]; aI += coef * pim[mp];
      }
      XrL[tid*PITCH + c] = aR;
      XiL[tid*PITCH + c] = aI;
    }
  }
  __syncthreads();

  // ---- psi_final = A * psi_enc : FP32 WMMA, M=256 in 16 tiles, N=16, K=256 ----
  {
    int lane = tid & 31, wv = tid >> 5;
    int lo = lane & 15, hi = lane >> 4;       // A/B fragment halves (K 0,1 | 2,3)
    for (int mt2 = 0; mt2 < 2; ++mt2) {
      int mbase = (wv*2 + mt2) * 16;
      int arow  = mbase + lo;
      v8f Dr = {0.f,0.f,0.f,0.f,0.f,0.f,0.f,0.f};
      v8f Di = {0.f,0.f,0.f,0.f,0.f,0.f,0.f,0.f};
      for (int k0 = 0; k0 < DIM; k0 += 4) {
        int ka = k0 + 2*hi;
        v2f a_r = *(const v2f*)(Ar + arow*DIM + ka);
        v2f a_i = *(const v2f*)(Ai + arow*DIM + ka);
        v2f b_r, b_i;
        b_r.x = XrL[ka*PITCH + lo];  b_r.y = XrL[(ka+1)*PITCH + lo];
        b_i.x = XiL[ka*PITCH + lo];  b_i.y = XiL[(ka+1)*PITCH + lo];
        v2f b_in = -b_i;                      // f32 WMMA has no operand NEG bits
        Dr = __builtin_amdgcn_wmma_f32_16x16x4_f32(false, a_r, false, b_r,  (short)0, Dr, false, false);
        Dr = __builtin_amdgcn_wmma_f32_16x16x4_f32(false, a_i, false, b_in, (short)0, Dr, false, false);
        Di = __builtin_amdgcn_wmma_f32_16x16x4_f32(false, a_r, false, b_i,  (short)0, Di, false, false);
        Di = __builtin_amdgcn_wmma_f32_16x16x4_f32(false, a_i, false, b_r,  (short)0, Di, false, false);
      }
      // C/D layout: VGPR r -> M = r + 8*hi, N = lo
      #pragma unroll
      for (int r = 0; r < 8; ++r) {
        int n = mbase + r + 8*hi;
        PL[n*PITCH + lo] = Dr[r]*Dr[r] + Di[r]*Di[r];
      }
    }
  }
  __syncthreads();

  // ---- <Z_j> = sum_n (+/-) p_n ; relu ----
  if (tid < 128) {
    int c = tid & 15, j = tid >> 4;
    float sum = 0.f;
    for (int n = 0; n < DIM; ++n) {
      float p = PL[n*PITCH + c];
      sum += ((n >> (7 - j)) & 1) ? -p : p;
    }
    ezs[c*8 + j] = fmaxf(sum, 0.f);
  }
  __syncthreads();

  if (tid < 16) {
    int g = base + tid;
    if (g < total) {
      int b = g / nseg, s = g % nseg;
      float v;
      if (LAYER == 1) {
        v = ezs[tid*8 + 0];                   // only channel 0 feeds layer 2
      } else {
        v = ezs[tid*8 + 0];
        #pragma unroll
        for (int j = 1; j < 8; ++j) v = fmaxf(v, ezs[tid*8 + j]);
      }
      out[b*nseg + s] = v;
    }
  }
}

// ============================================================================
// Dense: out[b] = sum_s m2[b,s]*dw[s] + db
// ============================================================================
__global__ void dense_kernel(const float* __restrict__ m2, const float* __restrict__ dw,
                             const float* __restrict__ db, float* __restrict__ out, int nseg) {
  __shared__ float red[256];
  int b = blockIdx.x, tid = threadIdx.x;
  float p = 0.f;
  for (int s = tid; s < nseg; s += 256) p += m2[b*nseg + s] * dw[s];
  red[tid] = p;
  __syncthreads();
  for (int off = 128; off > 0; off >>= 1) {
    if (tid < off) red[tid] += red[tid + off];
    __syncthreads();
  }
  if (tid == 0) out[b] = red[0] + db[0];
}

// ============================================================================
extern "C" void kernel_launch(void* const* d_in, const int* in_sizes, int n_in,
                              void* d_out, int out_size, void* d_ws, size_t ws_size,
                              hipStream_t stream) {
  (void)in_sizes; (void)n_in; (void)out_size; (void)ws_size;
  const float* x       = (const float*)d_in[0];   // (32,1024,1)
  const float* weights = (const float*)d_in[1];   // (3,8)
  const float* dense_w = (const float*)d_in[2];   // (1022,1)
  const float* dense_b = (const float*)d_in[3];   // (1,)
  float* out = (float*)d_out;                     // (32,1)

  const int B  = 32;
  const int L  = 1024;
  const int L1 = 1023;   // layer-1 segments
  const int L2 = 1022;   // layer-2 segments

  float* ws   = (float*)d_ws;
  float* Ar   = ws;                    // 256*256
  float* Ai   = Ar  + DIM*DIM;         // 256*256
  float* p0r  = Ai  + DIM*DIM;         // 256
  float* p0i  = p0r + DIM;             // 256
  float* y1   = p0i + DIM;             // 32*1023 (relu'd channel-0 of layer 1)
  float* m2   = y1  + B*L1;            // 32*1022 (channel-max of layer 2)

  precompute_kernel<<<DIM + 1, 256, 0, stream>>>(weights, Ar, Ai, p0r, p0i);
  quanv_kernel<1><<<(B*L1)/16, 256, 0, stream>>>(x,  L,  Ar, Ai, p0r, p0i, y1, B);   // 2046 WGs
  quanv_kernel<2><<<(B*L2)/16, 256, 0, stream>>>(y1, L1, Ar, Ai, p0r, p0i, m2, B);   // 2044 WGs
  dense_kernel<<<B, 256, 0, stream>>>(m2, dense_w, dense_b, out, L2);
}